// FusedMultiPool_68848325754933
// MI455X (gfx1250) — compile-verified
//
#include <hip/hip_runtime.h>

typedef float v4f __attribute__((ext_vector_type(4)));

namespace {
constexpr int kB = 32;
constexpr int kC = 256;
constexpr int kS = 128;
constexpr int kK = 8;
constexpr int kHW = 64 * 64;                       // one channel plane = 4096 floats = 16KB
constexpr int kThreads = 256;                      // 8 wave32s
constexpr int kWaves = kThreads / 32;
constexpr int kIters = (kHW / 4) / kThreads;       // 4 float4 positions per thread
}

// out[b,s,:,:] = max_k x[b, idx[s,k], :, :]
//
// One block per (b,s) output plane; blockIdx = b*S + s (batch-major) so the
// per-batch 4MB slice of x stays hot in the 192MB L2 (avg channel reuse = 4x).
//
// Data movement: all 32 gfx1250 async load-to-LDS ops are issued up front.
// Key trick: the async op's INST_OFFSET is added to BOTH the LDS and the
// global address, and our per-group stride is 4096B on both sides (256 thr *
// 16B global; 8ch * 32lane * 16B LDS). So 8 base address pairs, computed once,
// serve all 4 groups via offset:{0,4096,8192,12288} — zero per-iteration
// address VALU. Consume group j after s_wait_asynccnt {24,16,8,0} (async
// loads complete in order). Output via non-temporal b128 stores.
__global__ __launch_bounds__(kThreads)
void fused_multipool_gather_max(const float* __restrict__ x,
                                const int* __restrict__ idx,
                                float* __restrict__ out) {
  // stage[wave][group][k][lane] : 8 * 4 * 8 * 32 * 16B = 128KB per block
  __shared__ v4f stage[kWaves][kIters][kK][32];

  const int blk  = blockIdx.x;          // b*S + s
  const int s    = blk & (kS - 1);
  const int b    = blk >> 7;            // / kS
  const int t    = threadIdx.x;
  const int wave = t >> 5;
  const int lane = t & 31;

  const float* xb = x + (size_t)b * ((size_t)kC * kHW);
  const int*   ip = idx + s * kK;       // uniform -> scalar loads

  // Per-channel base addresses, computed once.
  unsigned long long ga[kK];            // lane global address (group 0)
  unsigned           la[kK];            // lane LDS address   (group 0)
#pragma unroll
  for (int k = 0; k < kK; ++k) {
    ga[k] = (unsigned long long)(const void*)(xb + (size_t)ip[k] * kHW + (size_t)t * 4);
    la[k] = (unsigned)(size_t)&stage[wave][0][k][lane];  // low 32 bits == LDS offset
  }

  // Issue all 32 async loads (ASYNCcnt is 6-bit; 32 outstanding is legal).
  // Each instruction moves 512B contiguous (32 lanes x 16B) from one channel.
#pragma unroll
  for (int j = 0; j < kIters; ++j) {
#pragma unroll
    for (int k = 0; k < kK; ++k) {
      if (j == 0)
        asm volatile("global_load_async_to_lds_b128 %0, %1, off"
                     :: "v"(la[k]), "v"(ga[k]) : "memory");
      else if (j == 1)
        asm volatile("global_load_async_to_lds_b128 %0, %1, off offset:4096"
                     :: "v"(la[k]), "v"(ga[k]) : "memory");
      else if (j == 2)
        asm volatile("global_load_async_to_lds_b128 %0, %1, off offset:8192"
                     :: "v"(la[k]), "v"(ga[k]) : "memory");
      else
        asm volatile("global_load_async_to_lds_b128 %0, %1, off offset:12288"
                     :: "v"(la[k]), "v"(ga[k]) : "memory");
    }
  }

  v4f* op = (v4f*)(out + (size_t)blk * kHW);

  // Consume group j once its 8 loads have retired (in-order completion).
#pragma unroll
  for (int j = 0; j < kIters; ++j) {
    if (j == 0)      asm volatile("s_wait_asynccnt 24" ::: "memory");
    else if (j == 1) asm volatile("s_wait_asynccnt 16" ::: "memory");
    else if (j == 2) asm volatile("s_wait_asynccnt 8"  ::: "memory");
    else             asm volatile("s_wait_asynccnt 0"  ::: "memory");

    v4f m = stage[wave][j][0][lane];
#pragma unroll
    for (int k = 1; k < kK; ++k) {
      v4f v = stage[wave][j][k][lane];
      m.x = fmaxf(m.x, v.x);
      m.y = fmaxf(m.y, v.y);
      m.z = fmaxf(m.z, v.z);
      m.w = fmaxf(m.w, v.w);
    }

    // Streaming output: non-temporal so 67MB of results don't evict x from L2.
    __builtin_nontemporal_store(m, op + t + j * kThreads);
  }
}

extern "C" void kernel_launch(void* const* d_in, const int* in_sizes, int n_in,
                              void* d_out, int out_size, void* d_ws, size_t ws_size,
                              hipStream_t stream) {
  (void)in_sizes; (void)n_in; (void)out_size; (void)d_ws; (void)ws_size;
  const float* x   = (const float*)d_in[0];       // (32,256,64,64) f32
  const int*   idx = (const int*)d_in[1];         // (128,8) i32
  float*       out = (float*)d_out;               // (32,128,64,64) f32

  dim3 grid(kB * kS);      // 4096 planes, batch-major for L2 locality
  dim3 block(kThreads);
  fused_multipool_gather_max<<<grid, block, 0, stream>>>(x, idx, out);
}